// WMMSE_21689584845489
// MI455X (gfx1250) — compile-verified
//
#include <hip/hip_runtime.h>

typedef float v2f __attribute__((ext_vector_type(2)));
typedef float v8f __attribute__((ext_vector_type(8)));

#define BB 4
#define UU 8
#define NR 8
#define NT 32
#define DD 2
#define LB 3                      // L = B-1
#define RATIO 1e-9f               // NOISE_POWER / SIGNAL_MAX_POWER

struct cplx { float re, im; };
__device__ __forceinline__ cplx cmul (cplx a, cplx b){ return {a.re*b.re - a.im*b.im, a.re*b.im + a.im*b.re}; }
__device__ __forceinline__ cplx cmulc(cplx a, cplx b){ return {a.re*b.re + a.im*b.im, a.im*b.re - a.re*b.im}; } // a*conj(b)
__device__ __forceinline__ cplx cadd (cplx a, cplx b){ return {a.re+b.re, a.im+b.im}; }
__device__ __forceinline__ cplx csub (cplx a, cplx b){ return {a.re-b.re, a.im-b.im}; }
__device__ __forceinline__ cplx cinv (cplx a){ float d = 1.0f/(a.re*a.re + a.im*a.im + 1e-30f); return {a.re*d, -a.im*d}; }

__device__ __forceinline__ v8f wmma4(v2f a, v2f b, v8f c) {
  // V_WMMA_F32_16X16X4_F32 : D = A(16x4) * B(4x16) + C
  return __builtin_amdgcn_wmma_f32_16x16x4_f32(false, a, false, b, (short)0, c, false, false);
}

__global__ __launch_bounds__(256) void wmmse_fused(
    const float* __restrict__ Hre, const float* __restrict__ Him,
    const float* __restrict__ Vre, const float* __restrict__ Vim,
    const float* __restrict__ wgt, float* __restrict__ out, int plane)
{
  extern __shared__ float smem[];
  float* sVr   = smem;           // 2048  V[b][u][t][d]
  float* sVi   = smem + 2048;    // 2048
  float* sHVr  = smem + 4096;    // 4096  HV_a[l][i][r][j*2+d]  (reused: W1 / P2)
  float* sHVi  = smem + 8192;    // 4096
  float* sH2r  = smem + 12288;   // 512   HV2[l][i][r][d]  (becomes Y after solve)
  float* sH2i  = smem + 12800;   // 512
  float* sUr   = smem + 13312;   // 2048  Umat[l][k][r][s]
  float* sUi   = smem + 15360;   // 2048
  float* sWmr  = smem + 17408;   // 128   Wm[l][k][2][2]
  float* sWmi  = smem + 17536;   // 128
  float* sTr   = smem + 17664;   // 2048  T[b][u][r][s]
  float* sTi   = smem + 19712;   // 2048
  float* sAr   = smem + 21760;   // 1024  A 32x32
  float* sAi   = smem + 22784;   // 1024
  float* sRr   = smem + 23808;   // 512   RHS 32x16
  float* sRi   = smem + 24320;   // 512
  float* sw    = smem + 24832;   // 8
  float* sScal = smem + 24840;   // 8  [0]=sum|V|^2  [1]=trace acc  [2,3]=pivot inv

  const int tid  = threadIdx.x;
  const int wave = tid >> 5;
  const int lane = tid & 31;
  const int half = lane >> 4;     // 0: lanes 0-15, 1: lanes 16-31
  const int lan  = lane & 15;
  const int n    = blockIdx.x;

  // ---------------- Phase 0: stage V, sum |V|^2, zero U ----------------
  if (tid < 8) { sw[tid] = wgt[tid]; sScal[tid] = 0.0f; }
  for (int i = tid; i < 2048; i += 256) { sUr[i] = 0.f; sUi[i] = 0.f; }
  __syncthreads();
  {
    float lsum = 0.f;
    const size_t vb = (size_t)n * 2048;
    for (int i = tid; i < 2048; i += 256) {
      float vr = Vre[vb + i], vi = Vim[vb + i];
      sVr[i] = vr; sVi[i] = vi; lsum += vr*vr + vi*vi;
    }
    atomicAdd(&sScal[0], lsum);
  }
  __syncthreads();

  // ---------------- Loop over BS index a: HV + Gram(Umat) ----------------
  for (int a = 0; a < BB; ++a) {
    // Phase A: HV_a[l,i,j] = H[n,a,l,i] @ V[n,a,j]   (WMMA 16x16x4, complex)
    for (int task = wave; task < 16; task += 8) {
      const int l  = task >> 2;
      const int ip = task & 3;                      // user pair -> rows 0..15
      const int uM = ip*2 + (lan >> 3);
      const int rM = lan & 7;
      const size_t hb = ((((size_t)n*BB + a)*BB + l)*UU + uM)*(NR*NT) + (size_t)rM*NT;
      const int jN = lan >> 1, ddN = lan & 1;       // column N = j*2+d
      const int vb = (a*UU + jN)*(NT*DD) + ddN;
      v8f accRe = {0,0,0,0,0,0,0,0};
      v8f accIm = {0,0,0,0,0,0,0,0};
#pragma unroll
      for (int kk = 0; kk < 8; ++kk) {
        const int t0 = kk*4 + half*2;
        v2f ar, ai, br, bi;
        ar.x = Hre[hb + t0]; ar.y = Hre[hb + t0 + 1];
        ai.x = Him[hb + t0]; ai.y = Him[hb + t0 + 1];
        br.x = sVr[vb + t0*DD]; br.y = sVr[vb + (t0+1)*DD];
        bi.x = sVi[vb + t0*DD]; bi.y = sVi[vb + (t0+1)*DD];
        v2f nai = { -ai.x, -ai.y };
        accRe = wmma4(ar,  br, accRe);    // +Hr*Vr
        accRe = wmma4(nai, bi, accRe);    // -Hi*Vi
        accIm = wmma4(ar,  bi, accIm);    // +Hr*Vi
        accIm = wmma4(ai,  br, accIm);    // +Hi*Vr
      }
#pragma unroll
      for (int v = 0; v < 8; ++v) {
        const int M  = v + half*8;
        const int iu = ip*2 + (M >> 3);
        const int rr = M & 7;
        const int idx = ((l*UU + iu)*NR + rr)*16 + lan;
        sHVr[idx] = accRe[v]; sHVi[idx] = accIm[v];
      }
    }
    __syncthreads();
    // diagonal HV2[n,a,i] = HV[a,a,i,i]
    if (tid < 128) {
      const int i = tid >> 4, r = (tid >> 1) & 7, d = tid & 1;
      const int src = ((a*UU + i)*NR + r)*16 + (i*DD + d);
      const int dst = ((a*UU + i)*NR + r)*DD + d;
      sH2r[dst] = sHVr[src]; sH2i[dst] = sHVi[src];
    }
    // Phase B: Umat[l,k] += X Xᴴ  (Gram via WMMA; pack k,k+1 in M and N)
    for (int task = wave; task < 16; task += 8) {
      const int l  = task >> 2;
      const int kp = task & 3;
      const int kA = kp*2 + (lan >> 3), rA = lan & 7;
      const int xb = ((l*UU + kA)*NR + rA)*16;
      v8f accRe = {0,0,0,0,0,0,0,0};
      v8f accIm = {0,0,0,0,0,0,0,0};
#pragma unroll
      for (int kk = 0; kk < 4; ++kk) {
        const int c0 = kk*4 + half*2;
        v2f ar, ai;
        ar.x = sHVr[xb + c0]; ar.y = sHVr[xb + c0 + 1];
        ai.x = sHVi[xb + c0]; ai.y = sHVi[xb + c0 + 1];
        v2f nai = { -ai.x, -ai.y };
        accRe = wmma4(ar, ar,  accRe);    // Xr Xr^T
        accRe = wmma4(ai, ai,  accRe);    // Xi Xi^T
        accIm = wmma4(ar, nai, accIm);    // -Xr Xi^T
        accIm = wmma4(ai, ar,  accIm);    // +Xi Xr^T
      }
#pragma unroll
      for (int v = 0; v < 8; ++v) {
        const int M = v + half*8, N = lan;
        if ((M >> 3) == (N >> 3)) {       // keep same-k diagonal 8x8 blocks
          const int k = kp*2 + (M >> 3);
          const int idx = (l*UU + k)*64 + (M & 7)*8 + (N & 7);
          sUr[idx] += accRe[v]; sUi[idx] += accIm[v];
        }
      }
    }
    __syncthreads();
  }

  // ---------------- beta_U regularizer on diagonal ----------------
  {
    const int l = tid >> 6, k = (tid >> 3) & 7, r = tid & 7;
    sUr[(l*UU + k)*64 + r*9] += RATIO * sScal[0];
  }
  __syncthreads();

  // ---------------- Phase C: 8x8 solves, Wm, T (one thread per (l,k)) ----------------
  if (tid < 32) {
    const int l = tid >> 3, k = tid & 7;
    const int ub = tid*64, hb2 = tid*16;
    float h2r[16], h2i[16];
    for (int x = 0; x < 16; ++x) { h2r[x] = sH2r[hb2+x]; h2i[x] = sH2i[hb2+x]; }
    // Gauss-Jordan on U | HV2 (in place; Y ends in sH2)
    for (int p = 0; p < 8; ++p) {
      cplx iv = cinv({ sUr[ub+p*8+p], sUi[ub+p*8+p] });
      for (int j = p; j < 8; ++j) {
        cplx x = cmul({ sUr[ub+p*8+j], sUi[ub+p*8+j] }, iv);
        sUr[ub+p*8+j] = x.re; sUi[ub+p*8+j] = x.im;
      }
      for (int d = 0; d < 2; ++d) {
        cplx x = cmul({ sH2r[hb2+p*2+d], sH2i[hb2+p*2+d] }, iv);
        sH2r[hb2+p*2+d] = x.re; sH2i[hb2+p*2+d] = x.im;
      }
      for (int r = 0; r < 8; ++r) {
        if (r == p) continue;
        cplx f = { sUr[ub+r*8+p], sUi[ub+r*8+p] };
        for (int j = p; j < 8; ++j) {
          cplx t = cmul(f, { sUr[ub+p*8+j], sUi[ub+p*8+j] });
          sUr[ub+r*8+j] -= t.re; sUi[ub+r*8+j] -= t.im;
        }
        for (int d = 0; d < 2; ++d) {
          cplx t = cmul(f, { sH2r[hb2+p*2+d], sH2i[hb2+p*2+d] });
          sH2r[hb2+r*2+d] -= t.re; sH2i[hb2+r*2+d] -= t.im;
        }
      }
    }
    // Wm = inv(I2 - Yᴴ HV2)
    cplx m00={1,0}, m01={0,0}, m10={0,0}, m11={1,0};
    for (int r = 0; r < 8; ++r) {
      cplx y0 = { sH2r[hb2+r*2+0], sH2i[hb2+r*2+0] };
      cplx y1 = { sH2r[hb2+r*2+1], sH2i[hb2+r*2+1] };
      cplx h0 = { h2r[r*2+0], h2i[r*2+0] };
      cplx h1 = { h2r[r*2+1], h2i[r*2+1] };
      m00 = csub(m00, cmulc(h0, y0)); m01 = csub(m01, cmulc(h1, y0));
      m10 = csub(m10, cmulc(h0, y1)); m11 = csub(m11, cmulc(h1, y1));
    }
    cplx id = cinv(csub(cmul(m00, m11), cmul(m01, m10)));
    cplx W00 = cmul(m11, id), W11 = cmul(m00, id);
    cplx W01 = cmul({-m01.re, -m01.im}, id), W10 = cmul({-m10.re, -m10.im}, id);
    sWmr[tid*4+0]=W00.re; sWmi[tid*4+0]=W00.im; sWmr[tid*4+1]=W01.re; sWmi[tid*4+1]=W01.im;
    sWmr[tid*4+2]=W10.re; sWmi[tid*4+2]=W10.im; sWmr[tid*4+3]=W11.re; sWmi[tid*4+3]=W11.im;
    // T = Y W Yᴴ
    cplx P[8][2];
    for (int r = 0; r < 8; ++r) {
      cplx y0 = { sH2r[hb2+r*2+0], sH2i[hb2+r*2+0] };
      cplx y1 = { sH2r[hb2+r*2+1], sH2i[hb2+r*2+1] };
      P[r][0] = cadd(cmul(y0, W00), cmul(y1, W10));
      P[r][1] = cadd(cmul(y0, W01), cmul(y1, W11));
    }
    float tr_acc = 0.f;
    for (int r = 0; r < 8; ++r)
      for (int s = 0; s < 8; ++s) {
        cplx y0 = { sH2r[hb2+s*2+0], sH2i[hb2+s*2+0] };
        cplx y1 = { sH2r[hb2+s*2+1], sH2i[hb2+s*2+1] };
        cplx t = cadd(cmulc(P[r][0], y0), cmulc(P[r][1], y1));
        sTr[tid*64 + r*8 + s] = t.re; sTi[tid*64 + r*8 + s] = t.im;
        if (r == s) tr_acc += t.re;
      }
    if (l == LB) atomicAdd(&sScal[1], sw[k] * tr_acc);
  }
  __syncthreads();

  // ---------------- Phase E: A = sum_{b,u} w Hᴴ T H  (WMMA) ----------------
  for (int i = tid; i < 1024; i += 256) { sAr[i] = 0.f; sAi[i] = 0.f; }
  __syncthreads();
  {
    v8f eRe[2][2], eIm[2][2];
#pragma unroll
    for (int mi = 0; mi < 2; ++mi)
#pragma unroll
      for (int ni = 0; ni < 2; ++ni) {
        v8f z = {0,0,0,0,0,0,0,0}; eRe[mi][ni] = z; eIm[mi][ni] = z;
      }
    float* sW1r = sHVr;  // reuse HV buffer: per-wave 8x32 scratch
    float* sW1i = sHVi;
    for (int pi = 0; pi < 4; ++pi) {
      const int pair = wave + pi*8;
      const int b = pair >> 3, u = pair & 7;
      const size_t hb = ((((size_t)n*BB + LB)*BB + b)*UU + u)*(NR*NT);
      // W1 = (w[u] T[b,u]) @ H[n,L,b,u]   (lane = column t)
      {
        const int t = lane;
        float hcr[8], hci[8];
        for (int s = 0; s < 8; ++s) { hcr[s] = Hre[hb + s*NT + t]; hci[s] = Him[hb + s*NT + t]; }
        const float wu = sw[u];
        const int tb = (b*UU + u)*64;
        for (int r = 0; r < 8; ++r) {
          float ar = 0.f, ai = 0.f;
          for (int s = 0; s < 8; ++s) {
            float tr_ = wu * sTr[tb + r*8 + s], ti_ = wu * sTi[tb + r*8 + s];
            ar += tr_*hcr[s] - ti_*hci[s];
            ai += tr_*hci[s] + ti_*hcr[s];
          }
          sW1r[wave*256 + r*NT + t] = ar;
          sW1i[wave*256 + r*NT + t] = ai;
        }
      }
      __syncthreads();
      // Apart += Hᴴ @ W1 : four 16x16 tiles, K=8
#pragma unroll
      for (int mi = 0; mi < 2; ++mi) {
        const int tIdx = mi*16 + lan;
#pragma unroll
        for (int kk = 0; kk < 2; ++kk) {
          const int r0 = kk*4 + half*2;
          v2f ar, ahi;
          ar.x  = Hre[hb + r0*NT + tIdx]; ar.y  = Hre[hb + (r0+1)*NT + tIdx];
          ahi.x = Him[hb + r0*NT + tIdx]; ahi.y = Him[hb + (r0+1)*NT + tIdx];
          v2f nahi = { -ahi.x, -ahi.y };
#pragma unroll
          for (int ni = 0; ni < 2; ++ni) {
            const int wb2 = wave*256 + ni*16 + lan;
            v2f bwr, bwi;
            bwr.x = sW1r[wb2 + r0*NT]; bwr.y = sW1r[wb2 + (r0+1)*NT];
            bwi.x = sW1i[wb2 + r0*NT]; bwi.y = sW1i[wb2 + (r0+1)*NT];
            eRe[mi][ni] = wmma4(ar,   bwr, eRe[mi][ni]);   // Hr^T W1r
            eRe[mi][ni] = wmma4(ahi,  bwi, eRe[mi][ni]);   // +Hi^T W1i
            eIm[mi][ni] = wmma4(ar,   bwi, eIm[mi][ni]);   // Hr^T W1i
            eIm[mi][ni] = wmma4(nahi, bwr, eIm[mi][ni]);   // -Hi^T W1r
          }
        }
      }
      __syncthreads();
    }
    // flush wave-partial tiles into A (LDS float atomics)
#pragma unroll
    for (int mi = 0; mi < 2; ++mi)
#pragma unroll
      for (int ni = 0; ni < 2; ++ni)
#pragma unroll
        for (int v = 0; v < 8; ++v) {
          const int row = mi*16 + v + half*8;
          const int col = ni*16 + lan;
          atomicAdd(&sAr[row*32 + col], eRe[mi][ni][v]);
          atomicAdd(&sAi[row*32 + col], eIm[mi][ni][v]);
        }
  }
  __syncthreads();
  if (tid < 32) sAr[tid*33] += (float)BB * RATIO * sScal[1];   // beta_u * I
  __syncthreads();

  // ---------------- Phase F: RHS, 32x32 solve, output ----------------
  {
    float* sP2r = sHVr;  // reuse again: P2[u][r][d]
    float* sP2i = sHVi;
    if (tid < 128) {
      const int u = tid >> 4, r = (tid >> 1) & 7, d = tid & 1;
      const int lk = LB*UU + u;
      cplx acc = {0,0};
      for (int e = 0; e < 2; ++e) {
        cplx y = { sH2r[lk*16 + r*2 + e], sH2i[lk*16 + r*2 + e] };
        cplx wm = { sWmr[lk*4 + e*2 + d], sWmi[lk*4 + e*2 + d] };
        acc = cadd(acc, cmul(y, wm));
      }
      sP2r[(u*8+r)*2+d] = acc.re; sP2i[(u*8+r)*2+d] = acc.im;
    }
    __syncthreads();
    {
      const int u = tid >> 5, t = tid & 31;
      const size_t hb = ((((size_t)n*BB + LB)*BB + LB)*UU + u)*(NR*NT);
      for (int d = 0; d < 2; ++d) {
        cplx acc = {0,0};
        for (int r = 0; r < 8; ++r) {
          cplx h = { Hre[hb + r*NT + t], Him[hb + r*NT + t] };
          cplx p = { sP2r[(u*8+r)*2+d], sP2i[(u*8+r)*2+d] };
          acc = cadd(acc, cmulc(p, h));            // conj(H)*P
        }
        sRr[t*16 + u*2 + d] = sw[u]*acc.re;
        sRi[t*16 + u*2 + d] = sw[u]*acc.im;
      }
    }
    __syncthreads();
    // Gauss-Jordan: A (32x32) | RHS (32x16)
    for (int p = 0; p < 32; ++p) {
      if (tid == 0) {
        cplx iv = cinv({ sAr[p*32+p], sAi[p*32+p] });
        sScal[2] = iv.re; sScal[3] = iv.im;
      }
      __syncthreads();
      cplx iv = { sScal[2], sScal[3] };
      if (tid < 48) {
        if (tid < 16) {
          cplx x = cmul({ sRr[p*16+tid], sRi[p*16+tid] }, iv);
          sRr[p*16+tid] = x.re; sRi[p*16+tid] = x.im;
        } else {
          const int j = p + (tid - 16) + 1;
          if (j < 32) {
            cplx x = cmul({ sAr[p*32+j], sAi[p*32+j] }, iv);
            sAr[p*32+j] = x.re; sAi[p*32+j] = x.im;
          }
        }
      }
      __syncthreads();
      for (int it = tid; it < 32*48; it += 256) {
        const int r = it / 48, jj = it % 48;
        if (r == p) continue;
        cplx f = { sAr[r*32+p], sAi[r*32+p] };
        if (jj < 16) {
          cplx t = cmul(f, { sRr[p*16+jj], sRi[p*16+jj] });
          sRr[r*16+jj] -= t.re; sRi[r*16+jj] -= t.im;
        } else {
          const int j = p + (jj - 16) + 1;
          if (j < 32) {
            cplx t = cmul(f, { sAr[p*32+j], sAi[p*32+j] });
            sAr[r*32+j] -= t.re; sAi[r*32+j] -= t.im;
          }
        }
      }
      __syncthreads();
    }
    // write V_L into out[:, n, L, u, t, d] (rest pre-zeroed)
    {
      const int u = tid >> 5, t = tid & 31;
      const size_t ob = (size_t)n*2048 + (size_t)LB*512 + (size_t)u*64 + (size_t)t*2;
      out[ob]             = sRr[t*16 + u*2 + 0];
      out[ob + 1]         = sRr[t*16 + u*2 + 1];
      out[(size_t)plane + ob]     = sRi[t*16 + u*2 + 0];
      out[(size_t)plane + ob + 1] = sRi[t*16 + u*2 + 1];
    }
  }
}

extern "C" void kernel_launch(void* const* d_in, const int* in_sizes, int n_in,
                              void* d_out, int out_size, void* d_ws, size_t ws_size,
                              hipStream_t stream) {
  (void)n_in; (void)d_ws; (void)ws_size;
  const float* Hre = (const float*)d_in[0];
  const float* Him = (const float*)d_in[1];
  const float* Vre = (const float*)d_in[2];
  const float* Vim = (const float*)d_in[3];
  const float* w   = (const float*)d_in[4];
  float* out = (float*)d_out;

  const int N = in_sizes[2] / (BB*UU*NT*DD);   // 1024
  const int plane = out_size / 2;              // N*B*U*Nt*d

  // b != L slices of the output are zero.
  (void)hipMemsetAsync(out, 0, (size_t)out_size * sizeof(float), stream);

  const size_t shbytes = 24848u * sizeof(float);   // ~97 KB of the 320 KB WGP LDS
  wmmse_fused<<<N, 256, shbytes, stream>>>(Hre, Him, Vre, Vim, w, out, plane);
}